// PlasticLinear_81879256531091
// MI455X (gfx1250) — compile-verified
//
#include <hip/hip_runtime.h>

// PlasticLinear: B=128, IN=512, OUT=512 (f32)
//   out[b,o]      = sum_i in[b,i]*(W[o,i] + PS[o,i]*H[b,o,i]) + bias[o]
//   Hnew[b,o,i]   = H + LR[o,i]*out*(in[b,i] - out*H[b,o,i])
//
// Bandwidth-bound (~271 MB min traffic). Design:
//   K1: out_ws = input @ W^T via V_WMMA_F32_16X16X4_F32 (one wave / 16x16 tile)
//   K2: one wave32 per (b,o) row: stream hebb row once (NT loads), reduce the
//       plastic dot term across the wave, fuse bias + plasticity update,
//       NT-store hebb_new. ps/lr/input stay regular-temporal (L2-resident).

#define B_   128
#define IN_  512
#define OUT_ 512

typedef float v2f __attribute__((ext_vector_type(2)));
typedef float v4f __attribute__((ext_vector_type(4)));
typedef float v8f __attribute__((ext_vector_type(8)));

// ---------------------------------------------------------------------------
// Kernel 1: out_ws[b,o] = sum_i input[b,i] * weight[o,i]   (FP32 WMMA GEMM)
// One wave per 16x16 (b,o) tile; K-loop in steps of 4 -> 128 WMMAs per wave.
// A-layout: lane half h=lane>>4, lr=lane&15: holds A[M=lr, K=k0+2h+{0,1}]
// B-layout (mirrored): holds B[K=k0+2h+{0,1}, N=lr] = weight[oBase+lr, k]
// D-layout: vgpr v -> D[M=v+8h, N=lr]
// ---------------------------------------------------------------------------
__global__ __launch_bounds__(128) void plastic_gemm_wmma(
    const float* __restrict__ input,   // (B, IN)
    const float* __restrict__ weight,  // (OUT, IN)
    float* __restrict__ out_ws)        // (B, OUT)
{
  const int lane  = threadIdx.x & 31;
  const int wave  = threadIdx.x >> 5;
  const int tile  = blockIdx.x * 4 + wave;      // 256 tiles = (128/16)*(512/16)
  const int tileB = tile >> 5;                  // OUT_/16 == 32 tiles per row
  const int tileO = tile & 31;
  const int bBase = tileB * 16;
  const int oBase = tileO * 16;
  const int half  = lane >> 4;
  const int lr    = lane & 15;

  const float* __restrict__ arow = input  + (size_t)(bBase + lr) * IN_;
  const float* __restrict__ brow = weight + (size_t)(oBase + lr) * IN_;

  v8f acc = {0.f, 0.f, 0.f, 0.f, 0.f, 0.f, 0.f, 0.f};

  #pragma unroll 4
  for (int k0 = 0; k0 < IN_; k0 += 4) {
    const int k = k0 + 2 * half;                // even -> 8B-aligned float2
    v2f a = *(const v2f*)(arow + k);
    v2f b = *(const v2f*)(brow + k);
    // 8 args: (neg_a, A, neg_b, B, c_mod, C, reuse_a, reuse_b)
    acc = __builtin_amdgcn_wmma_f32_16x16x4_f32(
        false, a, false, b, (short)0, acc, false, false);
  }

  float* __restrict__ dst =
      out_ws + (size_t)(bBase + 8 * half) * OUT_ + oBase + lr;
  #pragma unroll
  for (int v = 0; v < 8; ++v)
    dst[(size_t)v * OUT_] = acc[v];
}

// ---------------------------------------------------------------------------
// Kernel 2: one wave per (b,o) row (65536 rows). Lane l owns i = c*128+4l+t.
//   g   = sum_i in[i]*ps[i]*h[i]           (wave32 xor-shuffle reduction)
//   out = out_ws[row] + g ; d_out.out[row] = out + bias[o]
//   hn  = h + lr*out*(in - out*h)          (NT store)
// ---------------------------------------------------------------------------
__global__ __launch_bounds__(256) void plastic_update(
    const float* __restrict__ input,     // (B, IN)
    const float* __restrict__ hebb,      // (B, OUT, IN)
    const float* __restrict__ pscale,    // (OUT, IN)
    const float* __restrict__ plr,       // (OUT, IN)
    const float* __restrict__ bias,      // (OUT,)
    const float* __restrict__ out_ws,    // (B, OUT) from kernel 1
    float* __restrict__ out,             // (B, OUT)
    float* __restrict__ hebb_new)        // (B, OUT, IN)
{
  const int lane = threadIdx.x & 31;
  const int wave = threadIdx.x >> 5;
  const int row  = blockIdx.x * 8 + wave;        // 0 .. B_*OUT_-1
  const int b    = row >> 9;                     // / OUT_
  const int o    = row & (OUT_ - 1);

  const float* __restrict__ hrow = hebb   + (size_t)row * IN_;
  const float* __restrict__ prow = pscale + (size_t)o   * IN_;
  const float* __restrict__ lrow = plr    + (size_t)o   * IN_;
  const float* __restrict__ irow = input  + (size_t)b   * IN_;
  float*       __restrict__ nrow = hebb_new + (size_t)row * IN_;

  v4f h[4], in[4], pl[4];
  float g = 0.f;

  #pragma unroll
  for (int c = 0; c < 4; ++c) {
    const int idx = c * 128 + lane * 4;          // coalesced 512B per chunk
    h[c]  = __builtin_nontemporal_load((const v4f*)(hrow + idx));  // stream
    in[c] = *(const v4f*)(irow + idx);           // hot, keep temporal
    v4f p = *(const v4f*)(prow + idx);           // hot
    pl[c] = *(const v4f*)(lrow + idx);           // hot
    v4f t = in[c] * p * h[c];
    g += t.x + t.y + t.z + t.w;
  }

  // wave32 butterfly reduction -> every lane holds the full sum
  #pragma unroll
  for (int off = 16; off > 0; off >>= 1)
    g += __shfl_xor(g, off, 32);

  const float f = out_ws[row] + g;               // pre-bias activation
  if (lane == 0)
    out[row] = f + bias[o];

  #pragma unroll
  for (int c = 0; c < 4; ++c) {
    const int idx = c * 128 + lane * 4;
    v4f hn = h[c] + pl[c] * f * (in[c] - f * h[c]);
    __builtin_nontemporal_store(hn, (v4f*)(nrow + idx));
  }
}

// ---------------------------------------------------------------------------
extern "C" void kernel_launch(void* const* d_in, const int* in_sizes, int n_in,
                              void* d_out, int out_size, void* d_ws,
                              size_t ws_size, hipStream_t stream) {
  const float* input  = (const float*)d_in[0];   // (B, IN)
  const float* hebb   = (const float*)d_in[1];   // (B, OUT, IN)
  const float* weight = (const float*)d_in[2];   // (OUT, IN)
  const float* pscale = (const float*)d_in[3];   // (OUT, IN)
  const float* plr    = (const float*)d_in[4];   // (OUT, IN)
  const float* bias   = (const float*)d_in[5];   // (OUT,)

  float* out      = (float*)d_out;                       // B*OUT floats
  float* hebb_new = out + (size_t)B_ * OUT_;             // B*OUT*IN floats
  float* out_ws   = (float*)d_ws;                        // B*OUT floats scratch

  // K1: 256 tiles of 16x16, 4 waves/block -> 64 blocks.
  plastic_gemm_wmma<<<64, 128, 0, stream>>>(input, weight, out_ws);

  // K2: 65536 rows, 8 waves/block -> 8192 blocks.
  plastic_update<<<(B_ * OUT_) / 8, 256, 0, stream>>>(
      input, hebb, pscale, plr, bias, out_ws, out, hebb_new);
}